// PtModule_73667279061059
// MI455X (gfx1250) — compile-verified
//
#include <hip/hip_runtime.h>

// out[i,j] = x[i,j]*2 + 5 - (i+j)   for x: [8192, 4096] fp32, row-major.
// Memory-bound streaming kernel: 128-bit NT loads/stores, 4x ILP per thread.

typedef float v4f __attribute__((ext_vector_type(4)));

#define N_COLS      4096u
#define COL_SHIFT   12u          // log2(N_COLS)
#define COL_MASK    4095u
#define TPB         256u         // 8 waves (wave32) per block
#define VEC_PER_TH  4u           // 4 x 16B = 64 B per thread
#define VEC_PER_BLK (TPB * VEC_PER_TH)  // 1024 vec4 per block

__device__ __forceinline__ v4f compute(v4f a, unsigned int v) {
    // v = vec4 index; p = element index of component 0
    unsigned int p = v << 2;
    float i = (float)(p >> COL_SHIFT);
    float j = (float)(p & COL_MASK);
    float c = i + j;                      // exact (<= 12286)
    v4f r;
    r.x = a.x * 2.0f + 5.0f - c;          // matches reference rounding:
    r.y = a.y * 2.0f + 5.0f - (c + 1.0f); // 2*x exact, then one rounded add,
    r.z = a.z * 2.0f + 5.0f - (c + 2.0f); // one rounded sub
    r.w = a.w * 2.0f + 5.0f - (c + 3.0f);
    return r;
}

__global__ __launch_bounds__(TPB) void ewise_affine_idx_kernel(
    const v4f* __restrict__ x, v4f* __restrict__ out, unsigned int nvec) {

    const unsigned int base = blockIdx.x * VEC_PER_BLK + threadIdx.x;

    if (base + 3u * TPB < nvec) {
        // Fast path: 4 independent 128-bit NT loads in flight, then compute,
        // then 4 NT stores. Coalesced: consecutive lanes -> consecutive vec4.
        v4f a0 = __builtin_nontemporal_load(&x[base + 0u * TPB]);
        v4f a1 = __builtin_nontemporal_load(&x[base + 1u * TPB]);
        v4f a2 = __builtin_nontemporal_load(&x[base + 2u * TPB]);
        v4f a3 = __builtin_nontemporal_load(&x[base + 3u * TPB]);

        v4f r0 = compute(a0, base + 0u * TPB);
        v4f r1 = compute(a1, base + 1u * TPB);
        v4f r2 = compute(a2, base + 2u * TPB);
        v4f r3 = compute(a3, base + 3u * TPB);

        __builtin_nontemporal_store(r0, &out[base + 0u * TPB]);
        __builtin_nontemporal_store(r1, &out[base + 1u * TPB]);
        __builtin_nontemporal_store(r2, &out[base + 2u * TPB]);
        __builtin_nontemporal_store(r3, &out[base + 3u * TPB]);
    } else {
        // Tail path (unused for 8192x4096, kept for robustness).
        #pragma unroll
        for (unsigned int k = 0; k < VEC_PER_TH; ++k) {
            unsigned int v = base + k * TPB;
            if (v < nvec) {
                v4f a = __builtin_nontemporal_load(&x[v]);
                __builtin_nontemporal_store(compute(a, v), &out[v]);
            }
        }
    }
}

extern "C" void kernel_launch(void* const* d_in, const int* in_sizes, int n_in,
                              void* d_out, int out_size, void* d_ws, size_t ws_size,
                              hipStream_t stream) {
    (void)n_in; (void)d_ws; (void)ws_size; (void)in_sizes;

    const v4f* x = (const v4f*)d_in[0];
    v4f* out     = (v4f*)d_out;

    // out_size = 8192*4096 elements; 4 elements per vec4
    unsigned int nvec   = (unsigned int)(out_size / 4);
    unsigned int blocks = (nvec + VEC_PER_BLK - 1u) / VEC_PER_BLK;  // 8192

    ewise_affine_idx_kernel<<<blocks, TPB, 0, stream>>>(x, out, nvec);
}